// PaiNNInteraction_50397146251315
// MI455X (gfx1250) — compile-verified
//
#include <hip/hip_runtime.h>
#include <hip/hip_bf16.h>

#define N_ATOMS_C 25000
#define N_PAIRS_C 800000
#define FDIM 64
#define F3 192

typedef __attribute__((ext_vector_type(2))) float v2f;
typedef __attribute__((ext_vector_type(8))) float v8f;

// ---------------------------------------------------------------------------
// Kernel 1: initialize outputs  q_out = emb, mu_out = mu  (float4 vectorized)
// ---------------------------------------------------------------------------
__global__ __launch_bounds__(256) void painn_init(const float4* __restrict__ emb,
                                                  const float4* __restrict__ mu,
                                                  float4* __restrict__ q_out,
                                                  float4* __restrict__ mu_out,
                                                  int nq4, int nmu4) {
    int stride = gridDim.x * blockDim.x;
    for (int i = blockIdx.x * blockDim.x + threadIdx.x; i < nq4; i += stride)
        q_out[i] = emb[i];
    for (int i = blockIdx.x * blockDim.x + threadIdx.x; i < nmu4; i += stride)
        mu_out[i] = mu[i];
}

// ---------------------------------------------------------------------------
// Kernel 2: per-atom MLP with V_WMMA_F32_16X16X4_F32
//   H = silu(E @ W1 + b1)  -> LDS;  X = H @ W2 + b2 -> workspace
// Block = 256 threads (8 waves) handles 16 atoms (one M-tile).
// Lane mapping (wave32):
//   A frag (16x4 f32, 2 VGPRs):  a[v] = A[lane&15][k + (lane>>4)*2 + v]
//   B frag (4x16 f32, 2 VGPRs):  b[v] = B[k + (lane>>4)*2 + v][n0 + (lane&15)]
//   C/D   (16x16 f32, 8 VGPRs):  d[r] = D[r + (lane>>4)*8][n0 + (lane&15)]
// OOB rows handled branch-free by clamping the row index; stores guarded only
// in the single tail block. SiLU uses v_exp_f32 + v_rcp_f32 (no IEEE divide).
// ---------------------------------------------------------------------------
__global__ __launch_bounds__(256) void painn_mlp_wmma(const float* __restrict__ emb,  // (N,64)
                                                      const float* __restrict__ W1,   // (64,64)
                                                      const float* __restrict__ b1,   // (64)
                                                      const float* __restrict__ W2,   // (64,192)
                                                      const float* __restrict__ b2,   // (192)
                                                      float* __restrict__ x_out,      // (N,192)
                                                      int n_atoms) {
    __shared__ float Hs[16][FDIM + 4];   // +4 pad: conflict-free column reads

    const int lane = threadIdx.x & 31;
    const int wave = threadIdx.x >> 5;   // 0..7
    const int lrow = lane & 15;          // M-row (A) / N-col (B,C,D)
    const int lhi  = lane >> 4;          // K-pair select (A,B) / M-half (C,D)

    const int atom0 = blockIdx.x * 16;
    if (atom0 >= n_atoms) return;                  // block-uniform
    const bool fullTile = (atom0 + 16) <= n_atoms; // block-uniform

    // ---- Phase 1: waves 0..3 each compute one 16x16 tile of H ----
    if (wave < 4) {
        const int n0 = wave * 16;
        v8f acc;
        const float bias = b1[n0 + lrow];
#pragma unroll
        for (int r = 0; r < 8; ++r) acc[r] = bias;

        const int ar = min(atom0 + lrow, n_atoms - 1);         // branch-free clamp
        const float* erow = emb + (size_t)ar * FDIM + lhi * 2; // 8B aligned
        const float* wcol = W1 + n0 + lrow;

#pragma unroll
        for (int k = 0; k < FDIM; k += 4) {
            const float2 av = *(const float2*)(erow + k);      // global_load_b64
            v2f a, b;
            a[0] = av.x;
            a[1] = av.y;
            b[0] = wcol[(size_t)(k + lhi * 2 + 0) * FDIM];
            b[1] = wcol[(size_t)(k + lhi * 2 + 1) * FDIM];
            acc = __builtin_amdgcn_wmma_f32_16x16x4_f32(
                false, a, false, b, (short)0, acc, false, false);
        }
        // SiLU (fast rcp) then stage to LDS
#pragma unroll
        for (int r = 0; r < 8; ++r) {
            const int m = r + lhi * 8;
            const float h = acc[r];
            const float e = __expf(-h);                        // v_exp_f32
            Hs[m][n0 + lrow] = h * __builtin_amdgcn_rcpf(1.0f + e); // v_rcp_f32
        }
    }
    __syncthreads();

    // ---- Phase 2: 12 column tiles of X over 8 waves (wave-uniform loop) ----
    for (int t = wave; t < 12; t += 8) {
        const int n0 = t * 16;
        v8f acc;
        const float bias = b2[n0 + lrow];
#pragma unroll
        for (int r = 0; r < 8; ++r) acc[r] = bias;

        const float* hrow = &Hs[lrow][lhi * 2];   // 8B aligned (row stride 272B)
        const float* wcol = W2 + n0 + lrow;

#pragma unroll
        for (int k = 0; k < FDIM; k += 4) {
            const float2 hv = *(const float2*)(hrow + k);      // ds_load_b64
            v2f a, b;
            a[0] = hv.x;
            a[1] = hv.y;
            b[0] = wcol[(size_t)(k + lhi * 2 + 0) * F3];
            b[1] = wcol[(size_t)(k + lhi * 2 + 1) * F3];
            acc = __builtin_amdgcn_wmma_f32_16x16x4_f32(
                false, a, false, b, (short)0, acc, false, false);
        }

        float* xcol = x_out + (size_t)atom0 * F3 + n0 + lrow;
        if (fullTile) {
#pragma unroll
            for (int r = 0; r < 8; ++r)
                xcol[(size_t)(r + lhi * 8) * F3] = acc[r];
        } else {
#pragma unroll
            for (int r = 0; r < 8; ++r) {
                const int atom = atom0 + r + lhi * 8;
                if (atom < n_atoms)
                    xcol[(size_t)(r + lhi * 8) * F3] = acc[r];
            }
        }
    }
}

// ---------------------------------------------------------------------------
// Kernel 3: pair phase. One wave per pair; lane l owns features 2l, 2l+1.
// All loads are float2 (coalesced b64); scatter via global_atomic_add_f32.
// ---------------------------------------------------------------------------
__global__ __launch_bounds__(256) void painn_pairs(const float* __restrict__ Wij,    // (P,192)
                                                   const float* __restrict__ dir_ij, // (P,3)
                                                   const int* __restrict__ idx_i,
                                                   const int* __restrict__ idx_j,
                                                   const float* __restrict__ x,      // (N,192)
                                                   const float* __restrict__ mu,     // (N,3,64)
                                                   float* __restrict__ q_out,        // (N,64)
                                                   float* __restrict__ mu_out,       // (N,3,64)
                                                   int n_pairs) {
    const int lane   = threadIdx.x & 31;
    const int wavesPerBlock = blockDim.x >> 5;
    const int waveId = blockIdx.x * wavesPerBlock + (threadIdx.x >> 5);
    const int nWaves = gridDim.x * wavesPerBlock;
    const int f = lane * 2;

    for (int p = waveId; p < n_pairs; p += nWaves) {
        const int j = idx_j[p];
        const int i = idx_i[p];

        const float2* wp = (const float2*)(Wij + (size_t)p * F3);
        const float2* xp = (const float2*)(x   + (size_t)j * F3);
        const float2* mp = (const float2*)(mu  + (size_t)j * F3);

        // filter segments: [0:64)=dq, [64:128)=dmuR, [128:192)=dmumu
        const float2 wq = wp[lane];
        const float2 wR = wp[32 + lane];
        const float2 wM = wp[64 + lane];
        const float2 xq = xp[lane];
        const float2 xR = xp[32 + lane];
        const float2 xM = xp[64 + lane];

        const float d0 = dir_ij[(size_t)p * 3 + 0];
        const float d1 = dir_ij[(size_t)p * 3 + 1];
        const float d2 = dir_ij[(size_t)p * 3 + 2];

        const float dqx = wq.x * xq.x, dqy = wq.y * xq.y;
        const float aRx = wR.x * xR.x, aRy = wR.y * xR.y;
        const float aMx = wM.x * xM.x, aMy = wM.y * xM.y;

        const float2 m0 = mp[lane];
        const float2 m1 = mp[32 + lane];
        const float2 m2 = mp[64 + lane];

        float* qbase  = q_out  + (size_t)i * FDIM + f;
        float* mubase = mu_out + (size_t)i * F3 + f;

        atomicAdd(qbase + 0, dqx);
        atomicAdd(qbase + 1, dqy);

        atomicAdd(mubase + 0 * FDIM + 0, fmaf(aRx, d0, aMx * m0.x));
        atomicAdd(mubase + 0 * FDIM + 1, fmaf(aRy, d0, aMy * m0.y));
        atomicAdd(mubase + 1 * FDIM + 0, fmaf(aRx, d1, aMx * m1.x));
        atomicAdd(mubase + 1 * FDIM + 1, fmaf(aRy, d1, aMy * m1.y));
        atomicAdd(mubase + 2 * FDIM + 0, fmaf(aRx, d2, aMx * m2.x));
        atomicAdd(mubase + 2 * FDIM + 1, fmaf(aRy, d2, aMy * m2.y));
    }
}

// ---------------------------------------------------------------------------
extern "C" void kernel_launch(void* const* d_in, const int* in_sizes, int n_in,
                              void* d_out, int out_size, void* d_ws, size_t ws_size,
                              hipStream_t stream) {
    const float* emb    = (const float*)d_in[0];   // (25000,1,64)
    const float* mu     = (const float*)d_in[1];   // (25000,3,64)
    const float* Wij    = (const float*)d_in[2];   // (800000,1,192)
    const float* dir_ij = (const float*)d_in[3];   // (800000,3)
    const int*   idx_i  = (const int*)d_in[4];
    const int*   idx_j  = (const int*)d_in[5];
    const float* W1     = (const float*)d_in[6];   // (64,64)
    const float* b1     = (const float*)d_in[7];   // (64)
    const float* W2     = (const float*)d_in[8];   // (64,192)
    const float* b2     = (const float*)d_in[9];   // (192)

    const int n_atoms = N_ATOMS_C;
    const int n_pairs = N_PAIRS_C;

    float* q_out  = (float*)d_out;                          // (N,64)
    float* mu_out = (float*)d_out + (size_t)n_atoms * FDIM; // (N,3,64)
    float* x_ws   = (float*)d_ws;                           // (N,192) scratch

    // 1) outputs start as emb / mu
    {
        const int nq4  = (n_atoms * FDIM) / 4;
        const int nmu4 = (n_atoms * F3) / 4;
        painn_init<<<4096, 256, 0, stream>>>((const float4*)emb, (const float4*)mu,
                                             (float4*)q_out, (float4*)mu_out, nq4, nmu4);
    }

    // 2) per-atom MLP -> x in workspace (WMMA f32 16x16x4)
    {
        const int blocks = (n_atoms + 15) / 16;
        painn_mlp_wmma<<<blocks, 256, 0, stream>>>(emb, W1, b1, W2, b2, x_ws, n_atoms);
    }

    // 3) pair phase: one wave per pair, atomic scatter into outputs
    {
        const int wavesPerBlock = 256 / 32;
        const int blocks = (n_pairs + wavesPerBlock - 1) / wavesPerBlock;
        painn_pairs<<<blocks, 256, 0, stream>>>(Wij, dir_ij, idx_i, idx_j,
                                                x_ws, mu, q_out, mu_out, n_pairs);
    }
}